// EGARCH_15006615733133
// MI455X (gfx1250) — compile-verified
//
#include <hip/hip_runtime.h>
#include <math.h>

typedef __attribute__((ext_vector_type(2))) float v2f;
typedef __attribute__((ext_vector_type(8))) float v8f;

#define SQRT_2_OVER_PI 0.79788456080286535588f

// Tunables: wave chunk = TILES_PER_WAVE*256 elements; lookback = LB_TILES*256.
// beta^2048 underflows to 0 in fp32, so an 8-tile (2048-elt) lookback makes each
// wave's carry fp32-exact with NO inter-wave communication.
#define TILES_PER_WAVE 32
#define LB_TILES 8
#define RED_BLOCKS 1024

// ---------------- Pass 1a: deterministic partial reduction (sum, sumsq) ----------------
__global__ void __launch_bounds__(256) egarch_reduce(const float* __restrict__ r, int n,
                                                     double* __restrict__ part) {
    __shared__ double ls[256], ls2[256];
    double s = 0.0, s2 = 0.0;
    const size_t tid    = (size_t)blockIdx.x * 256 + threadIdx.x;
    const size_t stride = (size_t)gridDim.x * 256;
    const size_t nvec   = (size_t)n >> 2;
    const float4* r4    = (const float4*)r;
    for (size_t i = tid; i < nvec; i += stride) {        // b128 streaming loads
        float4 v = r4[i];
        double x0 = (double)v.x, x1 = (double)v.y, x2 = (double)v.z, x3 = (double)v.w;
        s += x0; s2 = fma(x0, x0, s2);
        s += x1; s2 = fma(x1, x1, s2);
        s += x2; s2 = fma(x2, x2, s2);
        s += x3; s2 = fma(x3, x3, s2);
    }
    for (size_t i = (nvec << 2) + tid; i < (size_t)n; i += stride) {   // tail
        double x = (double)r[i];
        s += x; s2 = fma(x, x, s2);
    }
    ls[threadIdx.x] = s; ls2[threadIdx.x] = s2;
    __syncthreads();
    for (int off = 128; off > 0; off >>= 1) {
        if ((int)threadIdx.x < off) {
            ls[threadIdx.x]  += ls[threadIdx.x + off];
            ls2[threadIdx.x] += ls2[threadIdx.x + off];
        }
        __syncthreads();
    }
    if (threadIdx.x == 0) {
        part[2 * blockIdx.x]     = ls[0];
        part[2 * blockIdx.x + 1] = ls2[0];
    }
}

// ---------------- Pass 1b: finalize var/std/log_h0, write out[0] and out[n] ----------------
__global__ void __launch_bounds__(256) egarch_finalize(const double* __restrict__ part,
                                                       int nparts, int n,
                                                       float* __restrict__ params,
                                                       float* __restrict__ out) {
    __shared__ double ls[256], ls2[256];
    double s = 0.0, s2 = 0.0;
    for (int i = threadIdx.x; i < nparts; i += 256) {
        s += part[2 * i]; s2 += part[2 * i + 1];
    }
    ls[threadIdx.x] = s; ls2[threadIdx.x] = s2;
    __syncthreads();
    for (int off = 128; off > 0; off >>= 1) {
        if ((int)threadIdx.x < off) {
            ls[threadIdx.x]  += ls[threadIdx.x + off];
            ls2[threadIdx.x] += ls2[threadIdx.x + off];
        }
        __syncthreads();
    }
    if (threadIdx.x == 0) {
        double N   = (double)n;
        double var = (ls2[0] - ls[0] * ls[0] / N) / (N - 1.0);
        double sd  = sqrt(var) + 1e-8;
        params[0]  = (float)(1.0 / sd);        // 1/std
        float lh0  = (float)log(var);          // log_h0
        params[1]  = lh0;
        float e = __expf(0.5f * lh0);
        out[0] = e;                            // exp(0.5*log_h[0])
        out[n] = e * e;                        // exp(log_h[0])
    }
}

// ---------------- Pass 2: WMMA-based parallel linear scan + outputs ----------------
// Scan variable: s[u] = beta*s[u-1] + c[u], s[-1] = log_h0, u in [0, n-2].
// Output: out[u+1] = exp(0.5*s[u]); out[n + u+1] = exp(s[u]).
// Tile (256 elts) = 16 segments x 16:  T = M_tri(beta) * C  via 4x V_WMMA_F32_16X16X4_F32.
__global__ void __launch_bounds__(256) egarch_scan(
    const float* __restrict__ ret, int n,
    const float* __restrict__ p_omega, const float* __restrict__ p_alpha,
    const float* __restrict__ p_beta,  const float* __restrict__ p_gamma,
    const float* __restrict__ params,  float* __restrict__ out) {

    const int lane = threadIdx.x & 31;
    const int col  = lane & 15;   // A-row (M) and B/D-column (N)
    const int hi   = lane >> 4;   // lane-half selects K pair / D row-half
    // readfirstlane -> wave id (and everything derived from it) is provably uniform,
    // so all tile-loop control flow compiles to SALU + s_cbranch, not EXEC juggling.
    const int wave = __builtin_amdgcn_readfirstlane(
        blockIdx.x * (blockDim.x >> 5) + (threadIdx.x >> 5));

    const long nscan       = (long)n - 1;
    const long tiles_total = (nscan + 255) / 256;
    const long first_tile  = (long)wave * TILES_PER_WAVE;
    if (first_tile >= tiles_total) return;
    long end_tile = first_tile + TILES_PER_WAVE;
    if (end_tile > tiles_total) end_tile = tiles_total;
    const long lb = (first_tile < LB_TILES) ? first_tile : (long)LB_TILES;
    const long t0 = first_tile - lb;
    // Only the very last tile of the array can read out of range.
    const long clamp_tile = (tiles_total * 256 > (long)n) ? (tiles_total - 1) : -1;

    const float beta    = p_beta[0];
    const float alpha   = p_alpha[0];
    const float gamma   = p_gamma[0];
    const float inv_std = params[0];
    const float lh0     = params[1];
    const float omega_a = p_omega[0] - alpha * SQRT_2_OVER_PI;

    // beta powers 0..15
    float bpow[16];
    bpow[0] = 1.0f;
#pragma unroll
    for (int i = 1; i < 16; ++i) bpow[i] = bpow[i - 1] * beta;
    const float q   = bpow[15] * beta;  // beta^16
    const float q2  = q * q;
    const float q4  = q2 * q2;
    const float q8  = q4 * q4;
    const float q16 = q8 * q8;          // beta^256 (per-tile carry factor)
    float qcol = 1.0f;
    for (int i = 0; i < col; ++i) qcol *= q;   // q^col

    // D-row carry factors: beta^(m+1), m = r + 8*hi
    float rowfac[8];
    const float b8 = bpow[8];
#pragma unroll
    for (int r = 0; r < 8; ++r) rowfac[r] = bpow[r] * beta * (hi ? b8 : 1.0f);

    // A-matrix (lower-triangular beta powers), split into 4 K-groups of 4.
    // Layout (32-bit A 16x4): M = lane&15; VGPR0 holds K=2*hi, VGPR1 holds K=2*hi+1.
    v2f Amat[4];
#pragma unroll
    for (int g = 0; g < 4; ++g) {
        const int kx = 4 * g + 2 * hi;
        Amat[g].x = (col >= kx)     ? bpow[col - kx]     : 0.0f;
        Amat[g].y = (col >= kx + 1) ? bpow[col - kx - 1] : 0.0f;
    }

    // Per-lane invariant offsets. Loads: u = col*16 + (4g + 2hi) -> base + off0 + 4*g
    // (the 4*g folds into the instruction immediate). Stores: 8 consecutive at orow.
    const int off0 = col * 16 + 2 * hi;
    const int orow = col * 16 + 8 * hi;

    float cw = (wave == 0) ? lh0 : 0.0f;   // carry entering current tile (s value)

    // Tile body helpers (inlined).
    auto compute = [&](const v2f (&in)[4]) {
        v8f a = {0.f, 0.f, 0.f, 0.f, 0.f, 0.f, 0.f, 0.f};
#pragma unroll
        for (int g = 0; g < 4; ++g) {
            const float zx = in[g].x * inv_std;
            const float zy = in[g].y * inv_std;
            v2f b;
            b.x = fmaf(alpha, fabsf(zx), fmaf(gamma, zx, omega_a));
            b.y = fmaf(alpha, fabsf(zy), fmaf(gamma, zy, omega_a));
            a = __builtin_amdgcn_wmma_f32_16x16x4_f32(
                false, Amat[g], false, b, (short)0, a, false, false);
        }
        return a;
    };
    // Weighted Hillis-Steele over the 16 columns (factor q = beta^16); updates cw,
    // returns the carry entering this lane's column.
    auto docarry = [&](const v8f& a) {
        float s = __shfl(a[7], 16 + col);   // T[15][col] lives in lane 16+col, VGPR 7
        float y;
        y = __shfl_up(s, 1); s = (col >= 1) ? fmaf(q,  y, s) : s;
        y = __shfl_up(s, 2); s = (col >= 2) ? fmaf(q2, y, s) : s;
        y = __shfl_up(s, 4); s = (col >= 4) ? fmaf(q4, y, s) : s;
        y = __shfl_up(s, 8); s = (col >= 8) ? fmaf(q8, y, s) : s;
        const float sp = __shfl_up(s, 1);
        const float ci = fmaf(qcol, cw, (col > 0) ? sp : 0.0f);
        const float s15 = __shfl(s, 15);
        cw = fmaf(q16, cw, s15);
        return ci;
    };
    auto load_tile = [&](const float* pb, long tile, v2f (&dst)[4]) {
        if (tile == clamp_tile) {
#pragma unroll
            for (int g = 0; g < 4; ++g) {
                long idx = tile * 256 + off0 + 4 * g;
                if (idx > (long)n - 2) idx = ((long)n - 2) & ~1L;
                dst[g] = *(const v2f*)(ret + idx);
            }
        } else {
#pragma unroll
            for (int g = 0; g < 4; ++g) dst[g] = *(const v2f*)(pb + off0 + 4 * g);
        }
    };

    v2f cur[4], nxt[4];
    const float* pnext = ret + t0 * 256;      // scalar base pointer, +256/tile
    load_tile(pnext, t0, cur);
#pragma unroll
    for (int g = 0; g < 4; ++g) nxt[g] = cur[g];
    pnext += 256;

    // ---- Lookback loop: establish carry, no stores, no guards ----
    for (long t = t0; t < first_tile; ++t) {
        load_tile(pnext, t + 1, nxt);         // next tile always exists here
        const v8f acc = compute(cur);
        (void)docarry(acc);
#pragma unroll
        for (int g = 0; g < 4; ++g) cur[g] = nxt[g];
        pnext += 256;
    }

    // ---- Main loop: always stores ----
    float* ob1 = out + first_tile * 256 + 1;  // scalar output bases, +256/tile
    float* ob2 = ob1 + n;
    long base  = first_tile * 256;
    for (long t = first_tile; t < end_tile; ++t) {
        if (t + 1 < end_tile) load_tile(pnext, t + 1, nxt);
        const v8f acc = compute(cur);
        const float carry_in = docarry(acc);

        if (base + 256 < (long)n) {           // fast path: whole tile in range
#pragma unroll
            for (int r = 0; r < 8; ++r) {
                const float lh = fmaf(rowfac[r], carry_in, acc[r]);
                const float e  = __expf(0.5f * lh);
                __builtin_nontemporal_store(e,     ob1 + orow + r);   // exp(0.5*log_h)
                __builtin_nontemporal_store(e * e, ob2 + orow + r);   // exp(log_h)
            }
        } else {                              // guarded final tile
#pragma unroll
            for (int r = 0; r < 8; ++r) {
                if (base + orow + r + 1 < (long)n) {
                    const float lh = fmaf(rowfac[r], carry_in, acc[r]);
                    const float e  = __expf(0.5f * lh);
                    ob1[orow + r] = e;
                    ob2[orow + r] = e * e;
                }
            }
        }
#pragma unroll
        for (int g = 0; g < 4; ++g) cur[g] = nxt[g];
        pnext += 256; ob1 += 256; ob2 += 256; base += 256;
    }
}

extern "C" void kernel_launch(void* const* d_in, const int* in_sizes, int n_in,
                              void* d_out, int out_size, void* d_ws, size_t ws_size,
                              hipStream_t stream) {
    const float* ret     = (const float*)d_in[0];
    const float* p_omega = (const float*)d_in[1];
    const float* p_alpha = (const float*)d_in[2];
    const float* p_beta  = (const float*)d_in[3];
    const float* p_gamma = (const float*)d_in[4];
    const int n = in_sizes[0];
    float* out = (float*)d_out;

    double* part  = (double*)d_ws;
    float* params = (float*)((char*)d_ws + (size_t)RED_BLOCKS * 2 * sizeof(double));

    egarch_reduce<<<RED_BLOCKS, 256, 0, stream>>>(ret, n, part);
    egarch_finalize<<<1, 256, 0, stream>>>(part, RED_BLOCKS, n, params, out);

    const long tiles = ((long)n - 1 + 255) / 256;
    const long waves = (tiles + TILES_PER_WAVE - 1) / TILES_PER_WAVE;
    const int blocks = (int)((waves + 7) / 8);   // 8 waves (256 threads) per block
    egarch_scan<<<blocks, 256, 0, stream>>>(ret, n, p_omega, p_alpha, p_beta, p_gamma,
                                            params, out);
}